// MCSPRLoss_39642548142549
// MI455X (gfx1250) — compile-verified
//
#include <hip/hip_runtime.h>
#include <hip/hip_bf16.h>

#define N_ROWS 8192
#define DY     1024
#define BDIM   256
#define TCTX   32
#define K_MINF 30.0f
#define TAUF   1e-4f
#define BETAF  0.9f

typedef __attribute__((ext_vector_type(2))) float v2f;
typedef __attribute__((ext_vector_type(8))) float v8f;

static __device__ __forceinline__ v8f wmma4(v2f a, v2f b, v8f c) {
    // V_WMMA_F32_16X16X4_F32 : D = A(16x4,f32) * B(4x16,f32) + C(16x16,f32)
    return __builtin_amdgcn_wmma_f32_16x16x4_f32(
        false, a, false, b, (short)0, c, false, false);
}

// ---------------------------------------------------------------------------
// Stage 1: Z[n,b] = sum_d Y[n,d] * Mp[b,d]     Z:[8192,256]
// One 32x32 output tile per wave: 2x2 sub-tiles of 16x16, 4 f32 accumulators.
// A frag: lane l holds A[l&15, k + 2*(l>>4) + {0,1}]  (float2, contiguous d)
// B frag: lane l holds B[k + 2*(l>>4) + {0,1}, l&15] = Mp[tn+(l&15), d..d+1]
// ---------------------------------------------------------------------------
__global__ __launch_bounds__(256) void k_gemm_Z(const float* __restrict__ Y,
                                                const float* __restrict__ Mp,
                                                float* __restrict__ Z) {
    const int lane = threadIdx.x & 31;
    const int wid  = (blockIdx.x * blockDim.x + threadIdx.x) >> 5;
    const int tm   = (wid >> 3) * 32;   // 256 M-tiles
    const int tn   = (wid & 7) * 32;    // 8 N-tiles
    const int lr   = lane & 15;
    const int lh   = lane >> 4;

    v8f c00 = {}, c01 = {}, c10 = {}, c11 = {};

    const float* Ya0 = Y  + (size_t)(tm + lr)      * DY + 2 * lh;
    const float* Ya1 = Y  + (size_t)(tm + 16 + lr) * DY + 2 * lh;
    const float* Bb0 = Mp + (size_t)(tn + lr)      * DY + 2 * lh;
    const float* Bb1 = Mp + (size_t)(tn + 16 + lr) * DY + 2 * lh;

#pragma unroll 4
    for (int k = 0; k < DY; k += 4) {
        v2f a0 = *(const v2f*)(Ya0 + k);
        v2f a1 = *(const v2f*)(Ya1 + k);
        v2f b0 = *(const v2f*)(Bb0 + k);
        v2f b1 = *(const v2f*)(Bb1 + k);
        c00 = wmma4(a0, b0, c00);
        c01 = wmma4(a0, b1, c01);
        c10 = wmma4(a1, b0, c10);
        c11 = wmma4(a1, b1, c11);
    }

#pragma unroll
    for (int r = 0; r < 8; ++r) {
        int rr = r + 8 * lh;   // C/D layout: VGPR r, lane-half lh -> row r+8*lh
        Z[(size_t)(tm + rr)      * BDIM + tn + lr]      = c00[r];
        Z[(size_t)(tm + rr)      * BDIM + tn + 16 + lr] = c01[r];
        Z[(size_t)(tm + 16 + rr) * BDIM + tn + lr]      = c10[r];
        Z[(size_t)(tm + 16 + rr) * BDIM + tn + 16 + lr] = c11[r];
    }
}

// ---------------------------------------------------------------------------
// Stage 2: mu[t,b] = sum_n w[n,t]*Z[n,b] / w_sum[t] ; eff_n[t]; zero loss acc
// ---------------------------------------------------------------------------
__global__ __launch_bounds__(256) void k_mu(const float* __restrict__ W,
                                            const float* __restrict__ Z,
                                            float* __restrict__ mu,
                                            float* __restrict__ effn,
                                            float* __restrict__ loss_sum) {
    const int t = blockIdx.x;
    const int b = threadIdx.x;
    float acc = 0.f, eff = 0.f;
    for (int n = 0; n < N_ROWS; ++n) {
        float wn = W[(size_t)n * TCTX + t];
        acc += wn * Z[(size_t)n * BDIM + b];
        eff += wn;
    }
    float wsum = eff > 0.f ? eff : 1.f;
    mu[t * BDIM + b] = acc / wsum;
    if (b == 0) { effn[t] = eff; loss_sum[t] = 0.f; }
}

// ---------------------------------------------------------------------------
// Stage 3: S[t,m,n] = sum_k w[k,t] * Z[k,m] * Z[k,n]   (raw; /w_sum later)
// S_t is symmetric: compute only upper-triangular 32x32 tiles (36 of 64) and
// mirror off-diagonal tiles on store -> ~44% less WMMA work.
// A[m,kk] = w[kk]*Z[kk,m] ; B[kk,n] = Z[kk,n] ; fragment loads hit contiguous
// 64B runs of Z rows across the 16 lanes of each half-wave.
// ---------------------------------------------------------------------------
__global__ __launch_bounds__(256) void k_gemm_S(const float* __restrict__ W,
                                                const float* __restrict__ Z,
                                                float* __restrict__ S) {
    const int lane = threadIdx.x & 31;
    const int wid  = (blockIdx.x * blockDim.x + threadIdx.x) >> 5;
    const int t    = wid / 36;          // 32 contexts x 36 upper-tri tiles
    int p          = wid % 36;
    int i = 0;                           // wave-uniform p -> (i,j), i<=j, 8x8
    while (p >= 8 - i) { p -= 8 - i; ++i; }
    const int j  = i + p;
    const int tm = i * 32;
    const int tn = j * 32;
    const int lr = lane & 15;
    const int lh = lane >> 4;

    v8f c00 = {}, c01 = {}, c10 = {}, c11 = {};

#pragma unroll 4
    for (int k = 0; k < N_ROWS; k += 4) {
        const int k0 = k + 2 * lh;
        const int k1 = k0 + 1;
        float w0 = W[(size_t)k0 * TCTX + t];
        float w1 = W[(size_t)k1 * TCTX + t];
        const float* z0 = Z + (size_t)k0 * BDIM;
        const float* z1 = Z + (size_t)k1 * BDIM;
        v2f a0 = { w0 * z0[tm + lr],      w1 * z1[tm + lr]      };
        v2f a1 = { w0 * z0[tm + 16 + lr], w1 * z1[tm + 16 + lr] };
        v2f b0 = { z0[tn + lr],           z1[tn + lr]           };
        v2f b1 = { z0[tn + 16 + lr],      z1[tn + 16 + lr]      };
        c00 = wmma4(a0, b0, c00);
        c01 = wmma4(a0, b1, c01);
        c10 = wmma4(a1, b0, c10);
        c11 = wmma4(a1, b1, c11);
    }

    float* St = S + (size_t)t * BDIM * BDIM;
#pragma unroll
    for (int r = 0; r < 8; ++r) {
        int rr = r + 8 * lh;
        St[(size_t)(tm + rr)      * BDIM + tn + lr]      = c00[r];
        St[(size_t)(tm + rr)      * BDIM + tn + 16 + lr] = c01[r];
        St[(size_t)(tm + 16 + rr) * BDIM + tn + lr]      = c10[r];
        St[(size_t)(tm + 16 + rr) * BDIM + tn + 16 + lr] = c11[r];
    }
    if (i != j) {   // mirror into the lower-triangular tile (wave-uniform branch)
#pragma unroll
        for (int r = 0; r < 8; ++r) {
            int rr = r + 8 * lh;
            St[(size_t)(tn + lr)      * BDIM + tm + rr]      = c00[r];
            St[(size_t)(tn + 16 + lr) * BDIM + tm + rr]      = c01[r];
            St[(size_t)(tn + lr)      * BDIM + tm + 16 + rr] = c10[r];
            St[(size_t)(tn + 16 + lr) * BDIM + tm + 16 + rr] = c11[r];
        }
    }
}

// ---------------------------------------------------------------------------
// Forward value of Sigma_unb[t,i,j] from raw S, mu, EMA state.
// ---------------------------------------------------------------------------
static __device__ __forceinline__ float sigma_unb_elem(float s_raw, float wsum,
                                                       float mui, float muj,
                                                       float ema, long long stp0) {
    float sb  = s_raw / wsum - mui * muj;
    float upd = (stp0 > 0) ? (BETAF * ema + (1.f - BETAF) * sb)
                           : ((1.f - BETAF) * sb);
    long long stp = stp0 + 1;
    float wg  = (stp > 1) ? (BETAF * upd + (1.f - BETAF) * sb)
                          : ((1.f - BETAF) * sb);
    float bc  = 1.f - powf(BETAF, (float)stp);
    return wg / bc;
}

__global__ __launch_bounds__(256) void k_std(const float* __restrict__ S,
                                             const float* __restrict__ mu,
                                             const float* __restrict__ effn,
                                             const float* __restrict__ ema,
                                             const long long* __restrict__ step,
                                             float* __restrict__ stdv) {
    const int t = blockIdx.x;
    const int i = threadIdx.x;
    float eff  = effn[t];
    float wsum = eff > 0.f ? eff : 1.f;
    size_t di = ((size_t)t * BDIM + i) * BDIM + i;
    float m = mu[t * BDIM + i];
    float d = sigma_unb_elem(S[di], wsum, m, m, ema[di], step[t]);
    stdv[t * BDIM + i] = sqrtf(d + TAUF);
}

// Stage 5: per-(t,chunk) partial sums of (clip(Sigma_unb/denom)-C_prior)^2
__global__ __launch_bounds__(256) void k_loss(const float* __restrict__ S,
                                              const float* __restrict__ mu,
                                              const float* __restrict__ effn,
                                              const float* __restrict__ ema,
                                              const long long* __restrict__ step,
                                              const float* __restrict__ stdv,
                                              const float* __restrict__ Cp,
                                              float* __restrict__ loss_sum) {
    const int t     = blockIdx.y;
    const int chunk = blockIdx.x;
    const int tid   = threadIdx.x;
    float eff  = effn[t];
    float wsum = eff > 0.f ? eff : 1.f;
    long long st = step[t];
    const float* St  = S   + (size_t)t * BDIM * BDIM;
    const float* Et  = ema + (size_t)t * BDIM * BDIM;
    const float* Ct  = Cp  + (size_t)t * BDIM * BDIM;
    const float* mut = mu   + t * BDIM;
    const float* sdt = stdv + t * BDIM;

    float acc = 0.f;
#pragma unroll 4
    for (int it = 0; it < 16; ++it) {
        int e = chunk * 4096 + it * 256 + tid;
        int i = e >> 8;
        int j = e & 255;
        float su = sigma_unb_elem(St[e], wsum, mut[i], mut[j], Et[e], st);
        float ch = su / (sdt[i] * sdt[j]);
        ch = fminf(1.f, fmaxf(-1.f, ch));
        float d = ch - Ct[e];
        acc += d * d;
    }

    __shared__ float red[256];
    red[tid] = acc;
    __syncthreads();
    for (int s = 128; s > 0; s >>= 1) {
        if (tid < s) red[tid] += red[tid + s];
        __syncthreads();
    }
    if (tid == 0) atomicAdd(&loss_sum[t], red[0]);
}

__global__ void k_final(const float* __restrict__ loss_sum,
                        const float* __restrict__ effn,
                        float* __restrict__ out) {
    __shared__ float tot[TCTX];
    __shared__ int   cnt[TCTX];
    int t = threadIdx.x;
    float eff = effn[t];
    bool active = eff >= K_MINF;
    float lt = loss_sum[t] / (eff * (float)(BDIM * BDIM));
    tot[t] = active ? lt : 0.f;
    cnt[t] = active ? 1 : 0;
    __syncthreads();
    if (t == 0) {
        float total = 0.f; int n = 0;
        for (int u = 0; u < TCTX; ++u) { total += tot[u]; n += cnt[u]; }
        float L = 0.1f * 1.0f * total / (float)TCTX / (1.0f - BETAF);
        out[0] = (n > 0) ? L : 0.f;
    }
}

extern "C" void kernel_launch(void* const* d_in, const int* in_sizes, int n_in,
                              void* d_out, int out_size, void* d_ws, size_t ws_size,
                              hipStream_t stream) {
    const float*     Y    = (const float*)d_in[0];      // [8192,1024]
    const float*     W    = (const float*)d_in[1];      // [8192,32]
    const float*     Mp   = (const float*)d_in[2];      // [256,1024]
    const float*     Cp   = (const float*)d_in[3];      // [32,256,256]
    const float*     ema  = (const float*)d_in[4];      // [32,256,256]
    const long long* stp  = (const long long*)d_in[5];  // [32] int64
    float* out = (float*)d_out;

    float* ws   = (float*)d_ws;
    float* Z    = ws;                         // 8192*256   = 2,097,152 f
    float* S    = Z + (size_t)N_ROWS * BDIM;  // 32*256*256 = 2,097,152 f
    float* mu   = S + (size_t)TCTX * BDIM * BDIM;   // 8192 f
    float* stdv = mu + TCTX * BDIM;                 // 8192 f
    float* effn = stdv + TCTX * BDIM;               // 32 f
    float* lsum = effn + TCTX;                      // 32 f

    // Z-GEMM: 2048 waves -> 256 blocks of 8 waves
    k_gemm_Z<<<256, 256, 0, stream>>>(Y, Mp, Z);
    k_mu<<<TCTX, 256, 0, stream>>>(W, Z, mu, effn, lsum);
    // S-GEMM (symmetric): 32 contexts x 36 upper-tri tiles = 1152 waves -> 144 blocks
    k_gemm_S<<<144, 256, 0, stream>>>(W, Z, S);
    k_std<<<TCTX, 256, 0, stream>>>(S, mu, effn, ema, stp, stdv);
    dim3 g5(16, TCTX);
    k_loss<<<g5, 256, 0, stream>>>(S, mu, effn, ema, stp, stdv, Cp, lsum);
    k_final<<<1, TCTX, 0, stream>>>(lsum, effn, out);
}